// DetectionOutput_83399674954431
// MI455X (gfx1250) — compile-verified
//
#include <hip/hip_runtime.h>
#include <hip/hip_bf16.h>

#define B_      16
#define P_      131072
#define NC      21
#define CM1     20
#define TOPK    400
#define KEEPK   200
#define TPAD    416          // 26 * 16
#define NWORD   13           // 416 / 32
#define NTASK   (B_ * CM1)   // 320
#define CAND    2048
#define NBUCK   4096
#define NEGV    (-1000000000.0f)
#define CONF_TH 0.01f
#define NMS_TH  0.45f

typedef __attribute__((ext_vector_type(2))) float v2f;
typedef __attribute__((ext_vector_type(8))) float v8f;

// ---------------------------------------------------------------- decode ----
__global__ void __launch_bounds__(256)
decode_kernel(const float* __restrict__ loc, const float* __restrict__ prior,
              float* __restrict__ dec) {
    int p = blockIdx.x * blockDim.x + threadIdx.x;
    int b = blockIdx.y;
    if (p >= P_) return;
    const float4 pb = ((const float4*)prior)[p];           // [xmin,ymin,xmax,ymax]
    float cx = (pb.x + pb.z) * 0.5f, cy = (pb.y + pb.w) * 0.5f;
    float w  = pb.z - pb.x,          h  = pb.w - pb.y;
    const float4 l = ((const float4*)(loc + (size_t)b * P_ * 4))[p];
    float x  = cx + l.x * 0.1f * w;
    float y  = cy + l.y * 0.1f * h;
    float nw = w * expf(l.z * 0.2f);
    float nh = h * expf(l.w * 0.2f);
    float4 o;
    o.x = x - nw * 0.5f; o.y = y - nh * 0.5f;
    o.z = o.x + nw;      o.w = o.y + nh;
    ((float4*)dec)[(size_t)b * P_ + p] = o;
}

// ----------------------------------------------- per-(b,class) top-400 ------
__global__ void __launch_bounds__(256)
topk_kernel(const float* __restrict__ conf, float* __restrict__ tops,
            int* __restrict__ topi) {
    __shared__ int   hist[NBUCK];
    __shared__ float cv[CAND];
    __shared__ int   ci[CAND];
    __shared__ int   s_cut, s_cnt;

    int t = blockIdx.x, b = t / CM1, c = t % CM1 + 1;
    int tid = threadIdx.x;

    for (int k = tid; k < NBUCK; k += 256) hist[k] = 0;
    if (tid == 0) s_cnt = 0;
    __syncthreads();

    const float* cp = conf + (size_t)b * P_ * NC + c;
    // Pass 1: histogram of thresholded scores (scores are in [0,1))
    for (int p = tid; p < P_; p += 256) {
        float v = cp[(size_t)p * NC];
        if (v > CONF_TH) {
            int bk = (int)(v * (float)NBUCK);
            bk = bk < 0 ? 0 : (bk > NBUCK - 1 ? NBUCK - 1 : bk);
            atomicAdd(&hist[bk], 1);
        }
    }
    __syncthreads();
    if (tid == 0) {
        int acc = 0, cut = 0;
        for (int k = NBUCK - 1; k >= 0; --k) { acc += hist[k]; if (acc >= TOPK) { cut = k; break; } }
        s_cut = cut;
    }
    __syncthreads();
    int cut = s_cut;
    // Pass 2: compact candidates in the winning buckets (conf re-read hits L2:
    // 176 MB < 192 MB global L2)
    for (int p = tid; p < P_; p += 256) {
        float v = cp[(size_t)p * NC];
        if (v > CONF_TH) {
            int bk = (int)(v * (float)NBUCK);
            bk = bk < 0 ? 0 : (bk > NBUCK - 1 ? NBUCK - 1 : bk);
            if (bk >= cut) {
                int pos = atomicAdd(&s_cnt, 1);
                if (pos < CAND) { cv[pos] = v; ci[pos] = p; }
            }
        }
    }
    __syncthreads();
    int n = s_cnt; if (n > CAND) n = CAND;
    for (int k = tid; k < CAND; k += 256)
        if (k >= n) { cv[k] = NEGV; ci[k] = 0x7FFFFFFF; }

    // Exact bitonic sort, descending by (value, index-ascending) == jax top_k order
    for (unsigned size = 2; size <= CAND; size <<= 1)
        for (unsigned stride = size >> 1; stride > 0; stride >>= 1) {
            __syncthreads();
            for (unsigned i = tid; i < CAND; i += 256) {
                unsigned j = i ^ stride;
                if (j > i) {
                    float av = cv[i], bv = cv[j];
                    int   ai = ci[i], bi = ci[j];
                    bool iFirst = (av > bv) || (av == bv && ai < bi);
                    bool descB  = ((i & size) == 0);
                    if (descB ? !iFirst : iFirst) {
                        cv[i] = bv; ci[i] = bi; cv[j] = av; ci[j] = ai;
                    }
                }
            }
        }
    __syncthreads();
    for (int k = tid; k < TOPK; k += 256) {
        tops[(size_t)t * TOPK + k] = cv[k];
        topi[(size_t)t * TOPK + k] = ci[k];
    }
}

// --------------------------------------- NMS with WMMA union-base tiles -----
__global__ void __launch_bounds__(256)
nms_kernel(const float* __restrict__ tops, const int* __restrict__ topi,
           const float* __restrict__ dec, float* __restrict__ sflat,
           float* __restrict__ bflat) {
    __shared__ float sb0[TPAD], sb1[TPAD], sb2[TPAD], sb3[TPAD];
    __shared__ float sar[TPAD], ssc[TPAD];
    __shared__ unsigned smask[TPAD * NWORD];
    __shared__ unsigned skeep[NWORD];

    int t = blockIdx.x, b = t / CM1, c = t % CM1 + 1;
    int tid = threadIdx.x;

    for (int k = tid; k < TPAD; k += 256) {
        float sv  = (k < TOPK) ? tops[(size_t)t * TOPK + k] : NEGV;
        int   idx = (k < TOPK) ? topi[(size_t)t * TOPK + k] : 0;
        float4 bx = make_float4(0.f, 0.f, 0.f, 0.f);
        if (sv > NEGV * 0.5f && idx >= 0 && idx < P_)
            bx = ((const float4*)dec)[(size_t)b * P_ + idx];
        sb0[k] = bx.x; sb1[k] = bx.y; sb2[k] = bx.z; sb3[k] = bx.w;
        sar[k] = (bx.z - bx.x) * (bx.w - bx.y);
        ssc[k] = sv;
    }
    for (int k = tid; k < TPAD * NWORD; k += 256) smask[k] = 0u;
    __syncthreads();

    int wave = tid >> 5, lane = tid & 31, half = lane >> 4, l16 = lane & 15;
    const int NT = TPAD / 16;               // 26
    const int NTRI = NT * (NT + 1) / 2;     // 351 upper-triangular tiles
    for (int tl = wave; tl < NTRI; tl += 8) {
        int tr = 0, rem = tl;
        while (rem >= NT - tr) { rem -= NT - tr; ++tr; }
        int tc = tr + rem;
        // A (16x4 fp32): VGPR0=K0/K2, VGPR1=K1/K3 -> lanes16-31 carry K2/K3 = 0
        v2f a, bb;
        if (half == 0) { a = (v2f){sar[tr * 16 + l16], 1.0f};
                         bb = (v2f){1.0f, sar[tc * 16 + l16]}; }
        else           { a = (v2f){0.f, 0.f}; bb = (v2f){0.f, 0.f}; }
        v8f acc = {};
        // C[m][n] = area_m*1 + 1*area_n  (union base) on the matrix core
        acc = __builtin_amdgcn_wmma_f32_16x16x4_f32(
                  false, a, false, bb, (short)0, acc, false, false);
        int n = tc * 16 + l16;
        float nx1 = sb0[n], ny1 = sb1[n], nx2 = sb2[n], ny2 = sb3[n];
#pragma unroll
        for (int v = 0; v < 8; ++v) {
            int m = tr * 16 + v + half * 8;   // C/D layout: VGPRv = rows v / v+8
            float ix1 = fmaxf(sb0[m], nx1);
            float iy1 = fmaxf(sb1[m], ny1);
            float ix2 = fminf(sb2[m], nx2);
            float iy2 = fminf(sb3[m], ny2);
            float inter = fmaxf(ix2 - ix1, 0.f) * fmaxf(iy2 - iy1, 0.f);
            float uni   = fmaxf(acc[v] - inter, 1e-9f);
            // inter/uni > TH  <=>  inter > TH*uni   (uni > 0) -- no fp divide
            if (n > m && n < TOPK && m < TOPK && inter > NMS_TH * uni)
                atomicOr(&smask[m * NWORD + (n >> 5)], 1u << (n & 31));
        }
    }
    __syncthreads();

    if (tid == 0) {
        for (int w = 0; w < NWORD; ++w) {
            unsigned kw = 0u;
            for (int bit = 0; bit < 32; ++bit) {
                int j = w * 32 + bit;
                if (j < TPAD && ssc[j] > NEGV * 0.5f) kw |= 1u << bit;
            }
            skeep[w] = kw;
        }
        for (int i = 0; i < TOPK; ++i)            // greedy suppression
            if (skeep[i >> 5] & (1u << (i & 31)))
                for (int w = 0; w < NWORD; ++w)
                    skeep[w] &= ~smask[i * NWORD + w];
        int cnt = 0;                              // cumsum(keep) <= 200 cap
        for (int j = 0; j < TOPK; ++j) {
            unsigned bm = 1u << (j & 31);
            if (skeep[j >> 5] & bm) { if (++cnt > KEEPK) skeep[j >> 5] &= ~bm; }
        }
    }
    __syncthreads();

    size_t base = (size_t)b * (CM1 * TOPK) + (size_t)(c - 1) * TOPK;
    for (int k = tid; k < TOPK; k += 256) {
        bool kept = (skeep[k >> 5] >> (k & 31)) & 1u;
        sflat[base + k] = kept ? ssc[k] : NEGV;
        float4 o; o.x = sb0[k]; o.y = sb1[k]; o.z = sb2[k]; o.w = sb3[k];
        ((float4*)bflat)[base + k] = o;
    }
}

// ------------------------------------------------ per-batch top-200 merge ---
__global__ void __launch_bounds__(1024)
final_kernel(const float* __restrict__ sflat, const float* __restrict__ bflat,
             float* __restrict__ out) {
    const int N = 8192, NR = CM1 * TOPK; // 8000 candidates
    __shared__ float fv[8192];
    __shared__ int   fi[8192];
    int b = blockIdx.x, tid = threadIdx.x, nt = blockDim.x;
    for (int k = tid; k < N; k += nt) {
        if (k < NR) { fv[k] = sflat[(size_t)b * NR + k]; fi[k] = k; }
        else        { fv[k] = NEGV; fi[k] = 0x7FFFFFFF; }
    }
    for (unsigned size = 2; size <= (unsigned)N; size <<= 1)
        for (unsigned stride = size >> 1; stride > 0; stride >>= 1) {
            __syncthreads();
            for (unsigned i = tid; i < (unsigned)N; i += nt) {
                unsigned j = i ^ stride;
                if (j > i) {
                    float av = fv[i], bv = fv[j];
                    int   ai = fi[i], bi = fi[j];
                    bool iFirst = (av > bv) || (av == bv && ai < bi);
                    bool descB  = ((i & size) == 0);
                    if (descB ? !iFirst : iFirst) {
                        fv[i] = bv; fi[i] = bi; fv[j] = av; fi[j] = ai;
                    }
                }
            }
        }
    __syncthreads();
    for (int k = tid; k < KEEPK; k += nt) {
        float v = fv[k]; int sel = fi[k];
        bool ok = (v > NEGV * 0.5f) && sel < NR;
        float4 bx = ok ? ((const float4*)bflat)[(size_t)b * NR + sel]
                       : make_float4(0.f, 0.f, 0.f, 0.f);
        float* row = out + ((size_t)b * KEEPK + k) * 7;
        row[0] = ok ? (float)b : 0.f;
        row[1] = ok ? (float)(sel / TOPK + 1) : 0.f;
        row[2] = ok ? v : 0.f;
        row[3] = bx.x; row[4] = bx.y; row[5] = bx.z; row[6] = bx.w;
    }
}

// ---------------------------------------------------------------- launch ----
extern "C" void kernel_launch(void* const* d_in, const int* in_sizes, int n_in,
                              void* d_out, int out_size, void* d_ws, size_t ws_size,
                              hipStream_t stream) {
    const float* loc   = (const float*)d_in[0];
    const float* conf  = (const float*)d_in[1];
    const float* prior = (const float*)d_in[2];
    float* out = (float*)d_out;

    char*  ws  = (char*)d_ws;
    size_t off = 0;
    float* dec   = (float*)(ws + off); off += (size_t)B_ * P_ * 4 * sizeof(float);
    float* tops  = (float*)(ws + off); off += (size_t)NTASK * TOPK * sizeof(float);
    int*   topi  = (int*)  (ws + off); off += (size_t)NTASK * TOPK * sizeof(int);
    float* sflat = (float*)(ws + off); off += (size_t)B_ * CM1 * TOPK * sizeof(float);
    float* bflat = (float*)(ws + off); off += (size_t)B_ * CM1 * TOPK * 4 * sizeof(float);
    (void)off; (void)ws_size; (void)in_sizes; (void)n_in; (void)out_size;

    decode_kernel<<<dim3(P_ / 256, B_), 256, 0, stream>>>(loc, prior, dec);
    topk_kernel  <<<NTASK, 256, 0, stream>>>(conf, tops, topi);
    nms_kernel   <<<NTASK, 256, 0, stream>>>(tops, topi, dec, sflat, bflat);
    final_kernel <<<B_, 1024, 0, stream>>>(sflat, bflat, out);
}